// ConditionalAttentionMixture_53008486367498
// MI455X (gfx1250) — compile-verified
//
#include <hip/hip_runtime.h>
#include <hip/hip_bf16.h>
#include <math.h>

#define Bn   64
#define Tn   2048
#define Dn   256
#define HIDn 32
#define Mn   3
#define EPSf 1e-8f

typedef float v2f __attribute__((ext_vector_type(2)));
typedef float v8f __attribute__((ext_vector_type(8)));

// ---------------- workspace layout (float offsets) ----------------
#define OFF_S      ((size_t)0)                       // B*T scores
#define OFF_W      (OFF_S    + (size_t)Bn*Tn)        // B*T attn weights
#define OFF_XM     (OFF_W    + (size_t)Bn*Tn)        // B*T channel means
#define OFF_DENS   (OFF_XM   + (size_t)Bn*Tn)        // B*T mask density
#define OFF_COLSUM (OFF_DENS + (size_t)Bn*Tn)        // B*D column sums (zeroed)
#define OFF_ACCUM  (OFF_COLSUM + (size_t)Bn*Dn)      // B*8 scalar accums (zeroed)
#define OFF_QV     (OFF_ACCUM  + (size_t)Bn*8)       // B*8 quantile rank values
#define OFF_POWER  (OFF_QV     + (size_t)Bn*8)       // B*1025 spectrum
#define OFF_GRAM   (OFF_POWER  + (size_t)Bn*1025)    // B*D*D gram

// ---------------- helpers ----------------
__device__ inline float block_reduce_sum(float v, float* buf) {
  int tid = threadIdx.x;
  buf[tid] = v; __syncthreads();
  for (int s = (int)blockDim.x >> 1; s > 0; s >>= 1) {
    if (tid < s) buf[tid] += buf[tid + s];
    __syncthreads();
  }
  float r = buf[0]; __syncthreads();
  return r;
}

__device__ inline unsigned f2key(float f) {
  unsigned u = __float_as_uint(f);
  return (u & 0x80000000u) ? ~u : (u ^ 0x80000000u);
}
__device__ inline float key2f(unsigned k) {
  unsigned u = (k & 0x80000000u) ? (k ^ 0x80000000u) : ~k;
  return __uint_as_float(u);
}

// ---------------- K1: per-row fused pass over h ----------------
// xm, dens, attention score, colsum, per-b raw moments / autocorr / observed
__global__ __launch_bounds__(256)
void row_kernel(const float* __restrict__ h, const int* __restrict__ mask,
                const float* __restrict__ Wa1, const float* __restrict__ ba1,
                const float* __restrict__ Wa2, const float* __restrict__ ba2,
                float* __restrict__ s, float* __restrict__ xm,
                float* __restrict__ dens, float* __restrict__ colsum,
                float* __restrict__ accum) {
  int row = blockIdx.x;            // b*Tn + t
  int b = row >> 11, t = row & (Tn - 1);
  int tid = threadIdx.x;           // == d
  __shared__ float lrow[Dn];
  __shared__ float red[256];
  size_t base = (size_t)row << 8;
  float x = h[base + tid];
  lrow[tid] = x;
  float xn = (t < Tn - 1) ? h[base + Dn + tid] : 0.0f;
  float mk = (float)mask[base + tid];
  atomicAdd(&colsum[(b << 8) + tid], x);
  __syncthreads();
  float x2 = x * x;
  float sumx = block_reduce_sum(x, red);
  float summ = block_reduce_sum(mk, red);
  float s2   = block_reduce_sum(x2, red);
  float s3   = block_reduce_sum(x2 * x, red);
  float s4   = block_reduce_sum(x2 * x2, red);
  float ac   = block_reduce_sum(x * xn, red);
  if (tid == 0) {
    xm[row]   = sumx * (1.0f / (float)Dn);
    dens[row] = summ / ((float)Dn + EPSf);
    float* A = accum + b * 8;
    atomicAdd(A + 0, sumx); atomicAdd(A + 1, s2);
    atomicAdd(A + 2, s3);   atomicAdd(A + 3, s4);
    atomicAdd(A + 4, ac);   atomicAdd(A + 5, summ);
  }
  // wave 0: attention score  s = tanh(row @ Wa1 + ba1) @ Wa2 + ba2
  if (tid < 32) {
    float acc = ba1[tid];
    #pragma unroll 8
    for (int d = 0; d < Dn; ++d) acc += lrow[d] * Wa1[d * HIDn + tid];
    float tz = tanhf(acc) * Wa2[tid];
    for (int off = 16; off; off >>= 1) tz += __shfl_down(tz, off, 32);
    if (tid == 0) s[row] = tz + ba2[0];
  }
}

// ---------------- K2: softmax over T per b ----------------
__global__ __launch_bounds__(256)
void softmax_kernel(const float* __restrict__ s, float* __restrict__ w) {
  int b = blockIdx.x, tid = threadIdx.x;
  __shared__ float red[256];
  const float* sb = s + b * Tn;
  float mx = -3.4e38f;
  for (int t = tid; t < Tn; t += 256) mx = fmaxf(mx, sb[t]);
  red[tid] = mx; __syncthreads();
  for (int r = 128; r; r >>= 1) { if (tid < r) red[tid] = fmaxf(red[tid], red[tid + r]); __syncthreads(); }
  float M = red[0]; __syncthreads();
  float sum = 0.f;
  for (int t = tid; t < Tn; t += 256) sum += expf(sb[t] - M);
  float S = block_reduce_sum(sum, red);
  for (int t = tid; t < Tn; t += 256) w[b * Tn + t] = expf(sb[t] - M) / S;
}

// ---------------- K3: attention pooling ----------------
__global__ __launch_bounds__(256)
void pool_kernel(const float* __restrict__ h, const float* __restrict__ w,
                 float* __restrict__ hp) {
  int b = blockIdx.x, d = threadIdx.x;
  const float* hb = h + ((size_t)b << 19);
  const float* wb = w + b * Tn;
  float acc = 0.f;
  #pragma unroll 4
  for (int t = 0; t < Tn; ++t) acc += hb[((size_t)t << 8) + d] * wb[t];
  hp[b * Dn + d] = acc;
}

// ---------------- K4: routing MLP + alpha ----------------
__global__ __launch_bounds__(32)
void routing_kernel(const float* __restrict__ hp,
                    const float* __restrict__ Wr1, const float* __restrict__ br1,
                    const float* __restrict__ Wr2, const float* __restrict__ br2,
                    const float* __restrict__ Wr3, const float* __restrict__ br3,
                    const float* __restrict__ temperature, float* __restrict__ alpha) {
  int b = blockIdx.x, tid = threadIdx.x;
  __shared__ float hrow[Dn]; __shared__ float z1[HIDn]; __shared__ float z2[HIDn];
  __shared__ float lg[Mn];
  for (int i = tid; i < Dn; i += 32) hrow[i] = hp[b * Dn + i];
  __syncthreads();
  float a = br1[tid];
  for (int d = 0; d < Dn; ++d) a += hrow[d] * Wr1[d * HIDn + tid];
  z1[tid] = fmaxf(a, 0.f); __syncthreads();
  float a2 = br2[tid];
  for (int k = 0; k < HIDn; ++k) a2 += z1[k] * Wr2[k * HIDn + tid];
  z2[tid] = fmaxf(a2, 0.f); __syncthreads();
  if (tid < Mn) {
    float l = br3[tid];
    for (int k = 0; k < HIDn; ++k) l += z2[k] * Wr3[k * Mn + tid];
    lg[tid] = l;
  }
  __syncthreads();
  if (tid == 0) {
    float temp = fminf(fmaxf(temperature[0], 0.5f), 2.0f);
    float l0 = lg[0] / temp, l1 = lg[1] / temp, l2 = lg[2] / temp;
    float mx = fmaxf(l0, fmaxf(l1, l2));
    float e0 = expf(l0 - mx), e1 = expf(l1 - mx), e2 = expf(l2 - mx);
    float ssum = e0 + e1 + e2;
    alpha[b * Mn + 0] = e0 / ssum; alpha[b * Mn + 1] = e1 / ssum; alpha[b * Mn + 2] = e2 / ssum;
  }
}

// ---------------- K5: naive rfft power spectrum of xm ----------------
__global__ __launch_bounds__(256)
void dft_kernel(const float* __restrict__ xm, float* __restrict__ power) {
  int idx = blockIdx.x * 256 + threadIdx.x;
  if (idx >= Bn * 1025) return;
  int b = idx / 1025, k = idx % 1025;
  const float* xr = xm + b * Tn;
  float re = 0.f, im = 0.f;
  const float wstep = -6.2831853071795864769f / (float)Tn;
  for (int t = 0; t < Tn; ++t) {
    float ang = wstep * (float)((t * k) & (Tn - 1));  // arg-reduced phase
    float si, co; sincosf(ang, &si, &co);
    float v = xr[t];
    re += v * co; im += v * si;
  }
  power[idx] = re * re + im * im;
}

// ---------------- K6: per-b scalar stats ----------------
__global__ __launch_bounds__(256)
void bstats_kernel(const float* __restrict__ xm, const float* __restrict__ dens,
                   const float* __restrict__ power, const float* __restrict__ accum,
                   float* __restrict__ stats) {
  int b = blockIdx.x, tid = threadIdx.x;
  __shared__ float red[256]; __shared__ float sval[256]; __shared__ int sidx[256];
  const float* X = xm + b * Tn;
  const float* DS = dens + b * Tn;
  const float* P = power + b * 1025;
  // trend
  float num = 0.f, den = 0.f;
  for (int t = tid; t < Tn; t += 256) {
    float tc = (float)t - (float)(Tn - 1) * 0.5f;
    num += tc * X[t]; den += tc * tc;
  }
  num = block_reduce_sum(num, red);
  den = block_reduce_sum(den, red);
  // xm mean (for zero crossings)
  float sx = 0.f;
  for (int t = tid; t < Tn; t += 256) sx += X[t];
  float xmean = block_reduce_sum(sx, red) / (float)Tn;
  // peaks / zc / roc
  float pk = 0.f, zc = 0.f, roc = 0.f;
  for (int t = tid; t < Tn - 1; t += 256) {
    float d1 = X[t + 1] - X[t];
    roc += fabsf(d1);
    if (t >= 1) { float d0 = X[t] - X[t - 1]; if (d1 * d0 < 0.f) pk += 1.f; }
    if ((X[t + 1] - xmean) * (X[t] - xmean) < 0.f) zc += 1.f;
  }
  pk  = block_reduce_sum(pk, red);
  zc  = block_reduce_sum(zc, red);
  roc = block_reduce_sum(roc, red);
  // density mean / var(ddof=1) / first-observed
  float ds = 0.f; int fo = Tn;
  for (int t = tid; t < Tn; t += 256) { float dv = DS[t]; ds += dv; if (dv > 0.f && t < fo) fo = t; }
  float dmean = block_reduce_sum(ds, red) / (float)Tn;
  float dvv = 0.f;
  for (int t = tid; t < Tn; t += 256) { float e = DS[t] - dmean; dvv += e * e; }
  float dvar = block_reduce_sum(dvv, red) / (float)(Tn - 1);
  sidx[tid] = fo; __syncthreads();
  for (int r = 128; r; r >>= 1) { if (tid < r) sidx[tid] = min(sidx[tid], sidx[tid + r]); __syncthreads(); }
  int first_t = sidx[0]; __syncthreads();
  // spectrum: argmax (first-max tie break) + entropy
  float lv = -3.4e38f; int li = 1 << 30; float psum = 0.f;
  for (int k = tid; k < 1025; k += 256) { float p = P[k]; psum += p; if (p > lv) { lv = p; li = k; } }
  psum = block_reduce_sum(psum, red);
  sval[tid] = lv; sidx[tid] = li; __syncthreads();
  for (int r = 128; r; r >>= 1) {
    if (tid < r) {
      if (sval[tid + r] > sval[tid] || (sval[tid + r] == sval[tid] && sidx[tid + r] < sidx[tid])) {
        sval[tid] = sval[tid + r]; sidx[tid] = sidx[tid + r];
      }
    }
    __syncthreads();
  }
  int kmax = sidx[0]; __syncthreads();
  float ent = 0.f;
  for (int k = tid; k < 1025; k += 256) { float pn = P[k] / (psum + EPSf); ent -= pn * logf(pn + EPSf); }
  ent = block_reduce_sum(ent, red);
  if (tid == 0) {
    const float* A = accum + b * 8;
    const float N = (float)Tn * (float)Dn;
    float mu = A[0] / N;
    float e2 = A[1] / N, e3 = A[2] / N, e4 = A[3] / N;
    float m2 = e2 - mu * mu;
    float m3 = e3 - 3.f * mu * e2 + 2.f * mu * mu * mu;
    float m4 = e4 - 4.f * mu * e3 + 6.f * mu * mu * e2 - 3.f * mu * mu * mu * mu;
    float* S = stats + b * 17;
    S[0]  = A[4] / ((float)(Tn - 1) * (float)Dn);
    S[1]  = num / (den + EPSf);
    S[2]  = (float)kmax / (float)(Tn / 2);
    S[3]  = ent;
    S[4]  = m3 / (powf(m2, 1.5f) + EPSf);
    S[5]  = m4 / (m2 * m2 + EPSf);
    S[9]  = pk / (float)(Tn - 2);
    S[10] = zc / (float)(Tn - 1);
    S[11] = roc / (float)(Tn - 1);
    S[12] = 1.0f - A[5] / (N + EPSf);
    S[13] = dvar;
    S[14] = (first_t < Tn) ? (float)first_t / (float)Tn : 0.0f;
  }
}

// ---------------- K7: gram = Xc^T Xc via V_WMMA_F32_16X16X4_F32 ----------------
// LDS-staged: each block owns a 16(d) x 128(e) output panel split across 8
// waves. K (=T) swept in chunks of 32: coalesced global -> transposed LDS
// tile (pad stride 34 -> 8B aligned, conflict-free), then ds_load_b64-fed
// WMMAs. A-panel fetched once per block instead of once per wave.
#define TCH   32            // K-chunk (t rows) per stage
#define PADT  34            // padded LDS stride (even -> b64 aligned)
#define BOFFA 0             // ldsA: 16 cols x TCH
#define BOFFB (16 * PADT)   // ldsB: 128 cols x TCH
__global__ __launch_bounds__(256)
void gram_kernel(const float* __restrict__ h, const float* __restrict__ colsum,
                 float* __restrict__ gram) {
  int b = blockIdx.y;
  int bx = blockIdx.x;                 // 0..31
  int tid = threadIdx.x;
  int wave = tid >> 5;                 // 0..7
  int lane = tid & 31;
  int d0     = (bx >> 1) << 4;         // A panel: 16 d-columns
  int e_base = (bx & 1) << 7;          // B panel: 128 e-columns
  int e0 = e_base + (wave << 4);       // this wave's 16 e-columns
  int m  = lane & 15;                  // A row / B-C-D col within tile
  int hi = lane >> 4;
  const float* hb = h + ((size_t)b << 19);
  __shared__ float lds[(16 + 128) * PADT];
  v8f c = {};
  const float* pa = lds + BOFFA + m * PADT;
  const float* pb = lds + BOFFB + ((wave << 4) + m) * PADT;
  int toff = 2 * hi;
  for (int tbase = 0; tbase < Tn; tbase += TCH) {
    // stage A panel: 16 x 32 (transposed, column-major by d)
    for (int r = tid; r < 16 * TCH; r += 256) {
      int mA = r & 15, tt = r >> 4;
      lds[BOFFA + mA * PADT + tt] = hb[((size_t)(tbase + tt) << 8) + d0 + mA];
    }
    // stage B panel: 128 x 32 (transposed, column-major by e)
    for (int r = tid; r < 128 * TCH; r += 256) {
      int cB = r & 127, tt = r >> 7;
      lds[BOFFB + cB * PADT + tt] = hb[((size_t)(tbase + tt) << 8) + e_base + cB];
    }
    __syncthreads();
    #pragma unroll
    for (int t0 = 0; t0 < TCH; t0 += 4) {
      v2f a  = *(const v2f*)(pa + t0 + toff);   // ds_load_b64
      v2f bf = *(const v2f*)(pb + t0 + toff);   // ds_load_b64
      c = __builtin_amdgcn_wmma_f32_16x16x4_f32(false, a, false, bf,
                                                (short)0, c, false, false);
    }
    __syncthreads();
  }
  // rank-1 mean correction + store: gram = X^T X - T * mu mu^T
  const float invT = 1.0f / (float)Tn;
  float mu_col = colsum[(b << 8) + e0 + m] * invT;
  #pragma unroll
  for (int v = 0; v < 8; ++v) {
    int rr = d0 + v + 8 * hi;
    float mu_row = colsum[(b << 8) + rr] * invT;
    gram[(((size_t)b << 8) + rr) * Dn + (e0 + m)] = c[v] - (float)Tn * mu_row * mu_col;
  }
}

// ---------------- K8: mean off-diag corr + power-iteration var_exp ----------------
__global__ __launch_bounds__(256)
void eig_kernel(const float* __restrict__ gram, float* __restrict__ stats) {
  int b = blockIdx.x, d = threadIdx.x;
  __shared__ float sh[Dn]; __shared__ float red[256];
  const float* G = gram + ((size_t)b << 16);
  const float* Grow = G + (size_t)d * Dn;
  float diag = Grow[d];
  float trace = block_reduce_sum(diag, red);
  sh[d] = sqrtf(diag / (float)(Tn - 1)); __syncthreads();
  float stdd = sh[d];
  float cs = 0.f;
  for (int e = 0; e < Dn; ++e) {
    if (e != d) {
      float cov = Grow[e] * (1.0f / ((float)(Tn - 1) + EPSf));
      cs += cov / (stdd * sh[e] + EPSf);
    }
  }
  float corr_sum = block_reduce_sum(cs, red);
  __syncthreads();
  sh[d] = 1.0f; __syncthreads();
  for (int it = 0; it < 48; ++it) {
    float y = 0.f;
    for (int e = 0; e < Dn; ++e) y += Grow[e] * sh[e];
    float n2 = block_reduce_sum(y * y, red);
    float inv = 1.0f / (sqrtf(n2) + 1e-30f);
    sh[d] = y * inv; __syncthreads();
  }
  float y = 0.f;
  for (int e = 0; e < Dn; ++e) y += Grow[e] * sh[e];
  float lam = block_reduce_sum(sh[d] * y, red);
  if (d == 0) {
    stats[b * 17 + 15] = corr_sum / ((float)Dn * (float)(Dn - 1) + EPSf);
    stats[b * 17 + 16] = lam / trace;
  }
}

// ---------------- K9: exact quantile ranks via 4-pass radix select ----------------
__global__ __launch_bounds__(256)
void select_kernel(const float* __restrict__ h, float* __restrict__ qvals) {
  int b = blockIdx.y, j = blockIdx.x;  // j in 0..5 -> {lo,hi} x {q25,q50,q75}
  int tid = threadIdx.x;
  const unsigned ranks[3] = {131071u, 262143u, 393215u};  // floor(q*(N-1))
  unsigned k = ranks[j >> 1] + (unsigned)(j & 1);
  const float* hb = h + ((size_t)b << 19);
  __shared__ unsigned hist[256];
  __shared__ unsigned sh_prefix; __shared__ unsigned sh_k;
  unsigned prefix = 0;
  for (int pass = 0; pass < 4; ++pass) {
    int shift = 24 - pass * 8;
    unsigned pmask = (pass == 0) ? 0u : (0xFFFFFFFFu << (shift + 8));
    hist[tid] = 0; __syncthreads();
    for (int i = tid; i < Tn * Dn; i += 256) {
      unsigned key = f2key(hb[i]);
      if ((key & pmask) == prefix) atomicAdd(&hist[(key >> shift) & 255u], 1u);
    }
    __syncthreads();
    if (tid == 0) {
      unsigned cum = 0, bsel = 0;
      for (int bin = 0; bin < 256; ++bin) {
        unsigned c = hist[bin];
        if (cum + c > k) { bsel = (unsigned)bin; break; }
        cum += c;
      }
      sh_prefix = prefix | (bsel << shift);
      sh_k = k - cum;
    }
    __syncthreads();
    prefix = sh_prefix; k = sh_k;
    __syncthreads();
  }
  if (tid == 0) qvals[b * 8 + j] = key2f(prefix);
}

// ---------------- K10: interpolate quantiles ----------------
__global__ __launch_bounds__(64)
void qfin_kernel(const float* __restrict__ qvals, float* __restrict__ stats) {
  int b = threadIdx.x;
  if (b >= Bn) return;
  const float frac[3] = {0.75f, 0.5f, 0.25f};  // pos - floor(pos) for q25/q50/q75
  for (int q = 0; q < 3; ++q) {
    float lo = qvals[b * 8 + 2 * q], hi = qvals[b * 8 + 2 * q + 1];
    stats[b * 17 + 6 + q] = lo + frac[q] * (hi - lo);
  }
}

// ---------------- launcher ----------------
extern "C" void kernel_launch(void* const* d_in, const int* in_sizes, int n_in,
                              void* d_out, int out_size, void* d_ws, size_t ws_size,
                              hipStream_t stream) {
  const float* h    = (const float*)d_in[0];
  const int*   mask = (const int*)d_in[1];
  const float* Wa1  = (const float*)d_in[2];
  const float* ba1  = (const float*)d_in[3];
  const float* Wa2  = (const float*)d_in[4];
  const float* ba2  = (const float*)d_in[5];
  const float* Wr1  = (const float*)d_in[6];
  const float* br1  = (const float*)d_in[7];
  const float* Wr2  = (const float*)d_in[8];
  const float* br2  = (const float*)d_in[9];
  const float* Wr3  = (const float*)d_in[10];
  const float* br3  = (const float*)d_in[11];
  const float* temperature = (const float*)d_in[12];

  float* out      = (float*)d_out;
  float* alpha    = out;                 // B*3
  float* h_pooled = out + Bn * Mn;       // B*256
  float* stats    = out + Bn * Mn + Bn * Dn;  // B*17

  float* ws     = (float*)d_ws;
  float* sbuf   = ws + OFF_S;
  float* wbuf   = ws + OFF_W;
  float* xmbuf  = ws + OFF_XM;
  float* dnsbuf = ws + OFF_DENS;
  float* colsum = ws + OFF_COLSUM;
  float* accum  = ws + OFF_ACCUM;
  float* qvals  = ws + OFF_QV;
  float* power  = ws + OFF_POWER;
  float* gram   = ws + OFF_GRAM;

  // zero the accumulated regions (colsum + accum are adjacent)
  hipMemsetAsync(colsum, 0, (size_t)(Bn * Dn + Bn * 8) * sizeof(float), stream);

  row_kernel<<<Bn * Tn, 256, 0, stream>>>(h, mask, Wa1, ba1, Wa2, ba2,
                                          sbuf, xmbuf, dnsbuf, colsum, accum);
  softmax_kernel<<<Bn, 256, 0, stream>>>(sbuf, wbuf);
  pool_kernel<<<Bn, 256, 0, stream>>>(h, wbuf, h_pooled);
  routing_kernel<<<Bn, 32, 0, stream>>>(h_pooled, Wr1, br1, Wr2, br2, Wr3, br3,
                                        temperature, alpha);
  dft_kernel<<<(Bn * 1025 + 255) / 256, 256, 0, stream>>>(xmbuf, power);
  bstats_kernel<<<Bn, 256, 0, stream>>>(xmbuf, dnsbuf, power, accum, stats);
  gram_kernel<<<dim3(32, Bn), 256, 0, stream>>>(h, colsum, gram);
  eig_kernel<<<Bn, 256, 0, stream>>>(gram, stats);
  select_kernel<<<dim3(6, Bn), 256, 0, stream>>>(h, qvals);
  qfin_kernel<<<1, 64, 0, stream>>>(qvals, stats);
}